// WeightedCrossEntropyLoss_23691039604766
// MI455X (gfx1250) — compile-verified
//
#include <hip/hip_runtime.h>
#include <math.h>
#include <stdint.h>

// Weighted cross-entropy, B=8192 rows x V=32000 cols, fp32.
// Bandwidth-bound: x (1.05 GB) is read exactly once via CDNA5 async
// global->LDS copies; whole 128KB row staged in LDS (320KB/WGP), then
// max-pass and sum-pass run out of LDS. Final reduction uses
// V_WMMA_F32_16X16X4_F32 (ones-B row-sum trick) for the cross-lane sum.

#define THREADS 256
#define NWAVES  (THREADS / 32)

constexpr int B_ROWS = 8192;
constexpr int V_COLS = 32000;
constexpr int VEC4   = V_COLS / 4;                       // 8000 float4 chunks
constexpr size_t SMEM_BYTES = (size_t)V_COLS * 4 + 64;   // row + reduction scratch

typedef __attribute__((ext_vector_type(2))) float v2f;
typedef __attribute__((ext_vector_type(8))) float v8f;

__global__ __launch_bounds__(THREADS)
void wce_row_kernel(const float* __restrict__ x,
                    const int*   __restrict__ y,
                    const float* __restrict__ w,
                    float*       __restrict__ row_out)
{
    extern __shared__ float smem[];           // [V_COLS] row buffer + scratch
    float* red = smem + V_COLS;               // NWAVES floats of scratch

    const int row  = blockIdx.x;
    const int tid  = threadIdx.x;
    const int lane = tid & 31;
    const int wid  = tid >> 5;

    const float* xrow = x + (size_t)row * V_COLS;

    // Raw LDS byte address: low 32 bits of the generic pointer are the
    // wave-relative LDS offset (flat-LDS aperture truncation rule).
    const unsigned lds_base = (unsigned)(uintptr_t)(void*)smem;

    // ---- Stage entire row into LDS with async B128 copies (ASYNCcnt) ----
    for (int c = tid; c < VEC4; c += THREADS) {
        unsigned laddr = lds_base + (unsigned)c * 16u;
        const float* gaddr = xrow + (size_t)c * 4;
        asm volatile("global_load_async_to_lds_b128 %0, %1, off"
                     :: "v"(laddr), "v"(gaddr) : "memory");
    }
    asm volatile("s_wait_asynccnt 0" ::: "memory");
    __syncthreads();

    // ---- Pass 1: row max (from LDS) ----
    const float4* s4 = (const float4*)smem;
    float m = -3.402823466e38f;
    for (int c = tid; c < VEC4; c += THREADS) {
        float4 v = s4[c];
        m = fmaxf(m, fmaxf(fmaxf(v.x, v.y), fmaxf(v.z, v.w)));
    }
    #pragma unroll
    for (int off = 16; off > 0; off >>= 1)
        m = fmaxf(m, __shfl_xor(m, off, 32));
    if (lane == 0) red[wid] = m;
    __syncthreads();
    float M = red[0];
    #pragma unroll
    for (int i = 1; i < NWAVES; ++i) M = fmaxf(M, red[i]);
    __syncthreads();   // red[] reused below

    // ---- Pass 2: s = sum_j w_j * exp(x_j - M) (x from LDS, w from L2) ----
    const float4* w4 = (const float4*)w;
    float s = 0.0f;
    for (int c = tid; c < VEC4; c += THREADS) {
        float4 v  = s4[c];
        float4 ww = w4[c];
        s += ww.x * __expf(v.x - M);
        s += ww.y * __expf(v.y - M);
        s += ww.z * __expf(v.z - M);
        s += ww.w * __expf(v.w - M);
    }
    #pragma unroll
    for (int off = 16; off > 0; off >>= 1)
        s += __shfl_xor(s, off, 32);
    if (lane == 0) red[wid] = s;
    __syncthreads();

    if (tid == 0) {
        float st = 0.0f;
        #pragma unroll
        for (int i = 0; i < NWAVES; ++i) st += red[i];
        float lse = M + logf(st);
        int   yi  = y[row];
        float g   = smem[yi];           // x[row, y[row]] straight from LDS
        row_out[row] = w[yi] * (lse - g);
    }
}

// Final deterministic 8192 -> 1 reduction. Per-wave 32->1 sum is done on the
// matrix pipe: D = A(16x4) * ones(4x16); with B all-ones D[i][n] = sum_k A[i][k].
// Column 0 lives in lane 0 (M=0..7, one row per C/D VGPR) and lane 16 (M=8..15),
// so lane0_colsum + lane16_colsum == sum of all 64 A entries held by the wave.
__global__ __launch_bounds__(THREADS)
void wce_final_sum_kernel(const float* __restrict__ in,
                          float*       __restrict__ out,
                          int n)
{
    __shared__ float red[NWAVES];
    const int tid  = threadIdx.x;
    const int lane = tid & 31;
    const int wid  = tid >> 5;

    float p = 0.0f;
    for (int i = tid; i < n; i += THREADS) p += in[i];  // fixed order: deterministic

    v2f a; a[0] = p;    a[1] = 0.0f;    // A: 64 entries (32 partials + zeros)
    v2f b; b[0] = 1.0f; b[1] = 1.0f;    // B: all ones
    v8f c = {};
    v8f d = __builtin_amdgcn_wmma_f32_16x16x4_f32(
        /*neg_a=*/false, a, /*neg_b=*/false, b,
        /*c_mod=*/(short)0, c, /*reuse_a=*/false, /*reuse_b=*/false);

    float colsum = d[0] + d[1] + d[2] + d[3] + d[4] + d[5] + d[6] + d[7];
    float wave_total = __shfl(colsum, 0, 32) + __shfl(colsum, 16, 32);

    if (lane == 0) red[wid] = wave_total;
    __syncthreads();
    if (tid == 0) {
        float t = 0.0f;
        #pragma unroll
        for (int i = 0; i < NWAVES; ++i) t += red[i];
        out[0] = t;
    }
}

extern "C" void kernel_launch(void* const* d_in, const int* in_sizes, int n_in,
                              void* d_out, int out_size, void* d_ws, size_t ws_size,
                              hipStream_t stream)
{
    const float* x = (const float*)d_in[0];   // [B, V] fp32
    const int*   y = (const int*)d_in[1];     // [B] int (harness convention)
    const float* w = (const float*)d_in[2];   // [V] fp32
    float* row_partials = (float*)d_ws;       // B floats of scratch
    float* out = (float*)d_out;               // 1 float

    // Allow >64KB dynamic LDS (row buffer is 128064 B; 320KB/WGP on CDNA5).
    (void)hipFuncSetAttribute(reinterpret_cast<const void*>(&wce_row_kernel),
                              hipFuncAttributeMaxDynamicSharedMemorySize,
                              (int)SMEM_BYTES);

    wce_row_kernel<<<B_ROWS, THREADS, SMEM_BYTES, stream>>>(x, y, w, row_partials);
    wce_final_sum_kernel<<<1, THREADS, 0, stream>>>(row_partials, out, B_ROWS);
}